// EmbeddingNet_35802847380183
// MI455X (gfx1250) — compile-verified
//
#include <hip/hip_runtime.h>
#include <hip/hip_bf16.h>

// ---------------------------------------------------------------------------
// Problem constants (match reference)
// ---------------------------------------------------------------------------
constexpr int  Bn   = 256;
constexpr int  Sn   = 2001;
constexpr int  Dn   = 128;
constexpr int  HALF = 1000;          // Sn // 2
constexpr int  LEAF = 2048;          // power-of-two >= Sn (segment tree leaves)

constexpr long FEA_OFF  = 0;
constexpr long POS_OFF  = (long)Bn * Sn * Dn;          // 65,568,768
constexpr long VIS_OFF  = 2 * POS_OFF;                 // 131,137,536
constexpr long TOP2_OFF = VIS_OFF + (long)Bn * Sn;     // 131,649,792

typedef float v2f __attribute__((ext_vector_type(2)));
typedef float v8f __attribute__((ext_vector_type(8)));

// ---------------------------------------------------------------------------
// Kernel 1: per-batch tour analysis, one workgroup (256 threads) per batch.
//   a) parallel list-ranking (pointer doubling, 11 passes) -> visit_index
//   b) push/pop interval construction per stack slot
//   c) max segment tree over pop-times; per-node O(log S) "two rightmost
//      active pushes" query reproduces jax.lax.top_k(stacks, 2) indices.
// Everything lives in LDS (9 * 2048 u32 = 72 KB of the 320 KB WGP pool).
// ---------------------------------------------------------------------------
__device__ inline int rightmost_gt(const unsigned* tr, int hi, unsigned t) {
  // rightmost leaf index u in [0, hi] with tr[LEAF+u] > t, else -1
  if (hi < 0) return -1;
  int l = LEAF, r = LEAF + hi;
  int candL[12]; int nl = 0;   // left-boundary cover nodes, left-to-right
  int candR[12]; int nr = 0;   // right-boundary cover nodes, right-to-left
  while (l <= r) {
    if (l & 1)    candL[nl++] = l++;
    if (!(r & 1)) candR[nr++] = r--;
    l >>= 1; r >>= 1;
  }
  int node = -1;
  for (int i = 0; i < nr; ++i)       if (tr[candR[i]] > t) { node = candR[i]; break; }
  if (node < 0)
    for (int i = nl - 1; i >= 0; --i) if (tr[candL[i]] > t) { node = candL[i]; break; }
  if (node < 0) return -1;
  while (node < LEAF) {
    int ri = 2 * node + 1;
    node = (tr[ri] > t) ? ri : (2 * node);
  }
  return node - LEAF;
}

__global__ void visit_scan_kernel(const int* __restrict__ sol,
                                  float* __restrict__ out) {
  extern __shared__ unsigned smem[];
  unsigned* succ    = smem;               // [LEAF]
  unsigned* pA      = succ    + LEAF;     // [LEAF]
  unsigned* wA      = pA      + LEAF;     // [LEAF]
  unsigned* pB      = wA      + LEAF;     // [LEAF]
  unsigned* wB      = pB      + LEAF;     // [LEAF]
  unsigned* rnk     = wB      + LEAF;     // [LEAF]
  unsigned* tree    = rnk     + LEAF;     // [2*LEAF] max-tree, leaves at [LEAF,2*LEAF)
  unsigned* pushpos = tree    + 2 * LEAF; // [LEAF]

  const int tid = threadIdx.x;
  const int NT  = blockDim.x;
  const int b   = blockIdx.x;
  const int* srow = sol + (long)b * Sn;

  // --- load successor row, init pointer-doubling state -----------------------
  for (int n = tid; n < LEAF; n += NT) {
    unsigned s = (n < Sn) ? (unsigned)srow[n] : 0u;
    succ[n] = s;
    bool root = (n == 0) || (n >= Sn);
    pA[n] = root ? 0u : s;
    wA[n] = root ? 0u : 1u;
  }
  __syncthreads();

  // --- list ranking: w[n] = distance n -> 0 along successors ----------------
  unsigned *ps = pA, *ws = wA, *pd = pB, *wd = wB;
  for (int it = 0; it < 11; ++it) {       // 2^11 = 2048 >= max distance 2000
    for (int n = tid; n < Sn; n += NT) {
      unsigned pp = ps[n];
      wd[n] = ws[n] + ws[pp];
      pd[n] = ps[pp];
    }
    __syncthreads();
    unsigned* tp_ = ps; ps = pd; pd = tp_;
    unsigned* tw_ = ws; ws = wd; wd = tw_;
  }
  // rank[n] = (Sn - dist) % Sn == visit step value (mod Sn)
  for (int n = tid; n < Sn; n += NT) {
    unsigned w = ws[n];
    rnk[n] = (w == 0u) ? 0u : (unsigned)Sn - w;
  }
  __syncthreads();

  // --- build push intervals: slot pos pushed at tp=rank[pos], popped at
  //     td=rank[pos+HALF]; active on [tp, te) with te = td>tp ? td : Sn+1 ----
  for (int u = tid; u < LEAF; u += NT) { tree[LEAF + u] = 0u; pushpos[u] = 0u; }
  __syncthreads();
  for (int pos = 1 + tid; pos <= HALF; pos += NT) {
    unsigned tp = rnk[pos];
    unsigned td = rnk[pos + HALF];
    unsigned te = (td > tp) ? td : (unsigned)(Sn + 1);
    tree[LEAF + tp]  = te;
    pushpos[tp]      = (unsigned)pos;
  }
  __syncthreads();
  // --- max segment tree bottom-up -------------------------------------------
  for (int half = LEAF >> 1; half >= 1; half >>= 1) {
    for (int i = half + tid; i < (half << 1); i += NT) {
      unsigned a = tree[2 * i], c = tree[2 * i + 1];
      tree[i] = a > c ? a : c;
    }
    __syncthreads();
  }

  // --- per-node queries: top2 at the step node n is visited -----------------
  for (int n = tid; n < Sn; n += NT) {
    unsigned rk = rnk[n];
    unsigned t  = (n == 0) ? (unsigned)Sn : rk;   // node 0 visited last (value Sn)
    int i1, i2;
    int u1 = rightmost_gt(tree, (int)t, t);
    if (u1 < 0)        { i1 = 0; i2 = 1; }        // 0 active: [0, 1]
    else {
      i1 = (int)pushpos[u1];
      int u2 = rightmost_gt(tree, u1 - 1, t);
      i2 = (u2 < 0) ? 0 : (int)pushpos[u2];       // 1 active: [pos, 0]
    }
    long base = (long)b * Sn + n;
    out[VIS_OFF + base]            = (float)rk;
    out[TOP2_OFF + base * 2 + 0]   = (float)i1;
    out[TOP2_OFF + base * 2 + 1]   = (float)i2;
  }
}

// ---------------------------------------------------------------------------
// Kernel 2: fea_emb = x[B*S, 2] @ W[128, 2]^T via V_WMMA_F32_16X16X4_F32.
// One wave per 16-row tile; K=2 padded to 4 (lanes 16..31 carry K=2,3 -> 0).
// 32-bit A 16x4 layout: lanes 0-15 hold (K0,K1) for M=lane; B 4x16 symmetric.
// ---------------------------------------------------------------------------
__global__ void fea_wmma_kernel(const float* __restrict__ x,
                                const float* __restrict__ W,
                                float* __restrict__ out) {
  const int  lane = threadIdx.x & 31;
  const int  wv   = threadIdx.x >> 5;
  const long tile = (long)blockIdx.x * 8 + wv;    // 16-row tile of the M dim
  const long m0   = tile * 16;
  if (m0 >= (long)Bn * Sn) return;                // grid sized exactly; wave-uniform

  v2f a = {0.f, 0.f};
  if (lane < 16) {                                // lanes 16..31: K=2,3 (zero pad)
    long row = m0 + lane;
    a.x = x[row * 2 + 0];
    a.y = x[row * 2 + 1];
  }

#pragma unroll
  for (int j = 0; j < 8; ++j) {                   // 8 column tiles of 16 -> N=128
    v2f bf = {0.f, 0.f};
    if (lane < 16) {
      int ncol = j * 16 + lane;
      bf.x = W[ncol * 2 + 0];                     // B[k=0][n] = W[n][0]
      bf.y = W[ncol * 2 + 1];                     // B[k=1][n] = W[n][1]
    }
    v8f c = {};
    c = __builtin_amdgcn_wmma_f32_16x16x4_f32(
        /*neg_a=*/false, a, /*neg_b=*/false, bf,
        /*c_mod=*/(short)0, c, /*reuse_a=*/false, /*reuse_b=*/false);

    // D layout: VGPR r holds M = r (lanes 0-15) / r+8 (lanes 16-31), N = lane%16
    const int  nc  = j * 16 + (lane & 15);
    const long mrb = m0 + (lane < 16 ? 0 : 8);
#pragma unroll
    for (int r = 0; r < 8; ++r)
      out[FEA_OFF + (mrb + r) * (long)Dn + nc] = c[r];
  }
}

// ---------------------------------------------------------------------------
// Kernel 3: pos_emb gather. pattern (1 MB) is L2-resident; one float4 per
// lane, 32 lanes cover one 512 B row -> fully coalesced load + store.
// ---------------------------------------------------------------------------
__global__ void pos_gather_kernel(const float* __restrict__ vis,
                                  const float* __restrict__ pattern,
                                  float* __restrict__ out) {
  long t = (long)blockIdx.x * blockDim.x + threadIdx.x;   // one float4 each
  long total = (long)Bn * Sn * (Dn / 4);
  if (t >= total) return;
  long pair = t >> 5;                 // Dn/4 == 32 quads per (b,s)
  int  q    = (int)(t & 31);
  int  vi   = (int)vis[pair];         // exact: value <= 2000
  const float4* src = (const float4*)pattern + (long)vi * (Dn / 4) + q;
  ((float4*)out)[t] = *src;
}

// ---------------------------------------------------------------------------
// Launch
// ---------------------------------------------------------------------------
extern "C" void kernel_launch(void* const* d_in, const int* in_sizes, int n_in,
                              void* d_out, int out_size, void* d_ws, size_t ws_size,
                              hipStream_t stream) {
  const float* x       = (const float*)d_in[0];   // [B,S,2]
  const int*   sol     = (const int*)  d_in[1];   // [B,S]
  const float* W       = (const float*)d_in[2];   // [D,2]
  const float* pattern = (const float*)d_in[3];   // [S,D]
  float*       out     = (float*)d_out;

  // 1) tour scan: one WG per batch, 72 KB dynamic LDS
  visit_scan_kernel<<<Bn, 256, 9 * LEAF * sizeof(unsigned), stream>>>(sol, out);

  // 2) fea_emb GEMM via WMMA: 32016 row-tiles, 8 waves/block -> 4002 blocks
  fea_wmma_kernel<<<4002, 256, 0, stream>>>(x, W, out);

  // 3) pos_emb gather (reads visit_index written by kernel 1; stream-ordered)
  pos_gather_kernel<<<64032, 256, 0, stream>>>(out + VIS_OFF, pattern,
                                               out + POS_OFF);
}